// StructuredPerceptron_8581344658203
// MI455X (gfx1250) — compile-verified
//
#include <hip/hip_runtime.h>
#include <hip/hip_bf16.h>

#define CHUNK 256

// ---------------------------------------------------------------------------
// CDNA5 async global->LDS copy (ASYNCcnt path, CDNA5 ISA).
// GVS mode: address is SGPR base plus 32-bit VGPR byte offset; LDS dst offset from a VGPR.
// Flat pointers into the LDS aperture carry the LDS byte offset in bits[31:0].
// ---------------------------------------------------------------------------
__device__ __forceinline__ void async_load_i32_to_lds(void* lds_dst,
                                                      unsigned global_byte_off,
                                                      const void* base) {
    unsigned lds_off = (unsigned)(unsigned long long)lds_dst;
    asm volatile("global_load_async_to_lds_b32 %0, %1, %2"
                 :: "v"(lds_off), "v"(global_byte_off), "s"(base)
                 : "memory");
}

__device__ __forceinline__ void wait_async_all() {
    asm volatile("s_wait_asynccnt 0" ::: "memory");
}

// ---------------------------------------------------------------------------
// Kernel 1: seq_len[b] = sum(mask[b, :])   (valid = pos < seq_len, NOT mask!)
// ---------------------------------------------------------------------------
__global__ void seqlen_kernel(const int* __restrict__ mask, int S,
                              int* __restrict__ seq_len) {
    __shared__ int red[256];
    const int b = blockIdx.x, tid = threadIdx.x;
    int acc = 0;
    for (int s = tid; s < S; s += 256) acc += mask[(size_t)b * S + s];
    red[tid] = acc;
    __syncthreads();
    for (int off = 128; off > 0; off >>= 1) {
        if (tid < off) red[tid] += red[tid + off];
        __syncthreads();
    }
    if (tid == 0) seq_len[b] = red[0];
}

// ---------------------------------------------------------------------------
// Kernel 2: per (batch, chunk) partial of (pred_score - gold_score).
// Stages tag/pred rows to LDS with async copies; gathers unary/binary diffs.
// ---------------------------------------------------------------------------
__global__ void score_kernel(const float* __restrict__ unary,
                             const float* __restrict__ binary,
                             const int* __restrict__ tags,
                             const int* __restrict__ pred,
                             const int* __restrict__ seq_len,
                             float* __restrict__ partials,
                             int S, int T, int CH) {
    __shared__ int   sh_tags[CHUNK + 2];
    __shared__ int   sh_pred[CHUNK + 2];
    __shared__ float red[CHUNK];

    const int b = blockIdx.y, c = blockIdx.x, tid = threadIdx.x;
    const int s0 = c * CHUNK;

    // Stage positions s0-1+i for i in [0, CHUNK] (extra leading element feeds
    // the transition term). Clamp pos<0 to 0; the value is never consumed.
    {
        int pos = s0 - 1 + tid;
        if (pos < 0) pos = 0;
        unsigned goff = (unsigned)((b * S + pos) * 4);
        async_load_i32_to_lds(&sh_tags[tid], goff, tags);
        async_load_i32_to_lds(&sh_pred[tid], goff, pred);
        if (tid == 0) {
            int pos2 = s0 + CHUNK - 1;
            unsigned goff2 = (unsigned)((b * S + pos2) * 4);
            async_load_i32_to_lds(&sh_tags[CHUNK], goff2, tags);
            async_load_i32_to_lds(&sh_pred[CHUNK], goff2, pred);
        }
    }

    // Warm L2 with the 1 MB binary table while the async copies are in flight
    // (global_prefetch_b8). Spread blocks across the table; 192 MB L2 keeps it.
    {
        unsigned fl  = (unsigned)(b * CH + c) * (unsigned)CHUNK + (unsigned)tid;
        unsigned n   = (unsigned)T * (unsigned)T;
        unsigned idx = (fl * 32u) % n;                 // one cacheline apart
        __builtin_prefetch(&binary[idx], 0, 1);
    }

    const int L = seq_len[b];

    wait_async_all();      // this wave's async copies landed in LDS
    __syncthreads();       // all waves' copies visible

    const int s = s0 + tid;
    float diff = 0.0f;
    if (s < S && s < L) {
        const int gt = sh_tags[tid + 1];
        const int pt = sh_pred[tid + 1];
        const float* row = unary + ((size_t)(b * S + s)) * (size_t)T;
        diff += row[pt] - row[gt];                     // unary gather diff
        if (s >= 1) {                                  // transition (s-1 -> s)
            const int gp = sh_tags[tid];
            const int pp = sh_pred[tid];
            diff += binary[(size_t)pp * T + pt] - binary[(size_t)gp * T + gt];
        }
    }

    red[tid] = diff;
    __syncthreads();
    for (int off = CHUNK / 2; off > 0; off >>= 1) {
        if (tid < off) red[tid] += red[tid + off];
        __syncthreads();
    }
    if (tid == 0) partials[b * CH + c] = red[0];
}

// ---------------------------------------------------------------------------
// Kernel 3: one wave32. Lane b: fixed-order sum of its chunk partials, relu,
// then fixed-order shfl_xor tree across 32 lanes -> d_out[0]. Deterministic.
// ---------------------------------------------------------------------------
__global__ void finalize_kernel(const float* __restrict__ partials,
                                int B, int CH, float* __restrict__ out) {
    const int lane = threadIdx.x;   // blockDim.x == 32 (one wave on CDNA5)
    float loss = 0.0f;
    if (lane < B) {
        float acc = 0.0f;
        for (int cidx = 0; cidx < CH; ++cidx) acc += partials[lane * CH + cidx];
        loss = fmaxf(acc, 0.0f);
    }
    for (int off = 16; off > 0; off >>= 1)
        loss += __shfl_xor(loss, off, 32);
    if (lane == 0) out[0] = loss;
}

// ---------------------------------------------------------------------------
extern "C" void kernel_launch(void* const* d_in, const int* in_sizes, int n_in,
                              void* d_out, int out_size, void* d_ws, size_t ws_size,
                              hipStream_t stream) {
    (void)n_in; (void)out_size; (void)ws_size;
    const float* unary  = (const float*)d_in[0];
    const float* binary = (const float*)d_in[1];
    const int*   tags   = (const int*)d_in[2];
    const int*   pred   = (const int*)d_in[3];
    const int*   mask   = (const int*)d_in[4];
    float* out = (float*)d_out;

    // T from binary (T x T); problem sizes from setup: B=32, S=2048, T=512.
    int T = 1;
    while ((long long)T * (long long)T < (long long)in_sizes[1]) T <<= 1;
    const int B  = 32;
    const int S  = in_sizes[2] / B;
    const int CH = (S + CHUNK - 1) / CHUNK;

    int*   seq_len  = (int*)d_ws;                          // B ints
    float* partials = (float*)((char*)d_ws + 256);         // B*CH floats

    seqlen_kernel<<<B, 256, 0, stream>>>(mask, S, seq_len);
    score_kernel<<<dim3(CH, B), CHUNK, 0, stream>>>(unary, binary, tags, pred,
                                                    seq_len, partials, S, T, CH);
    finalize_kernel<<<1, 32, 0, stream>>>(partials, B, CH, out);
}